// QuantumKernelLayer_65481071403005
// MI455X (gfx1250) — compile-verified
//
#include <hip/hip_runtime.h>

// Quantum layer: per-sample 2-qubit real-amplitude circuit.
// Memory-bound streaming kernel: 2 samples per thread -> all traffic is
// B128 non-temporal global loads/stores on gfx1250 (wave32).
// Trig via native v_sin_f32/v_cos_f32 (args are O(1) radians -> no range
// reduction needed), keeping VALU work far below the 23.3 TB/s HBM floor.

// Native clang vector type (NOT HIP_vector_type) so the nontemporal builtins accept it.
typedef float v4f __attribute__((ext_vector_type(4)));

__device__ __forceinline__ void qsample(float x0, float x1,
                                        float p0, float p1, float p2, float p3,
                                        float& z0, float& z1) {
    float sa, ca, sb, cb, s2, c2, s3, c3;
    __sincosf(0.5f * (x0 + p0), &sa, &ca);   // RY(x0)RY(p0) fused on q0
    __sincosf(0.5f * (x1 + p1), &sb, &cb);   // RY(x1)RY(p1) fused on q1
    __sincosf(0.5f * p2, &s2, &c2);
    __sincosf(0.5f * p3, &s3, &c3);

    // state[j1][j0] after the two fused RYs, then CX(ctrl=q0, tgt=q1):
    // t00 = s00, t01 = s11, t10 = s10, t11 = s01
    float t00 = cb * ca;
    float t01 = sb * sa;
    float t10 = sb * ca;
    float t11 = cb * sa;

    // RY(p2) acting on the last axis (q0): u[k,i] = sum_j R2[i,j] t[k,j]
    float u00 = c2 * t00 - s2 * t01;
    float u01 = s2 * t00 + c2 * t01;
    float u10 = c2 * t10 - s2 * t11;
    float u11 = s2 * t10 + c2 * t11;

    // RY(p3) acting on the middle axis (q1): v[i,k] = sum_j R3[i,j] u[j,k]
    float v00 = c3 * u00 - s3 * u10;
    float v01 = c3 * u01 - s3 * u11;
    float v10 = s3 * u00 + c3 * u10;
    float v11 = s3 * u01 + c3 * u11;

    float q00 = v00 * v00;
    float q01 = v01 * v01;
    float q10 = v10 * v10;
    float q11 = v11 * v11;

    z0 = (q00 + q10) - (q01 + q11);   // <Z> on qubit 0 (last axis)
    z1 = (q00 + q01) - (q10 + q11);   // <Z> on qubit 1 (middle axis)
}

__global__ __launch_bounds__(256) void quantum_layer_kernel(
    const v4f* __restrict__ x2,    // [nPairs]  : 2 samples of x[2]   per vec4
    const v4f* __restrict__ p4,    // [2*nPairs]: 1 sample  of par[4] per vec4
    v4f* __restrict__ out2,        // [nPairs]  : 2 samples of out[2] per vec4
    int nPairs) {
    int i = blockIdx.x * blockDim.x + threadIdx.x;
    if (i >= nPairs) return;

    // Streaming data, zero reuse -> non-temporal hints (TH=NT).
    v4f xv = __builtin_nontemporal_load(&x2[i]);          // x[2i], x[2i+1]
    v4f pa = __builtin_nontemporal_load(&p4[2 * i]);      // params[2i]
    v4f pb = __builtin_nontemporal_load(&p4[2 * i + 1]);  // params[2i+1]

    float oz0, oz1, oz2, oz3;
    qsample(xv.x, xv.y, pa.x, pa.y, pa.z, pa.w, oz0, oz1);
    qsample(xv.z, xv.w, pb.x, pb.y, pb.z, pb.w, oz2, oz3);

    v4f o;
    o.x = oz0;
    o.y = oz1;
    o.z = oz2;
    o.w = oz3;
    __builtin_nontemporal_store(o, &out2[i]);
}

// Tail kernel for odd B (not hit for B = 4,194,304, but keeps it general).
__global__ __launch_bounds__(64) void quantum_layer_tail_kernel(
    const float* __restrict__ x, const float* __restrict__ p,
    float* __restrict__ out, int start, int n) {
    int s = start + blockIdx.x * blockDim.x + threadIdx.x;
    if (s >= n) return;
    float z0, z1;
    qsample(x[2 * s], x[2 * s + 1],
            p[4 * s], p[4 * s + 1], p[4 * s + 2], p[4 * s + 3], z0, z1);
    out[2 * s] = z0;
    out[2 * s + 1] = z1;
}

extern "C" void kernel_launch(void* const* d_in, const int* in_sizes, int n_in,
                              void* d_out, int out_size, void* d_ws, size_t ws_size,
                              hipStream_t stream) {
    const float* x = (const float*)d_in[0];   // [B,2] float32
    const float* p = (const float*)d_in[1];   // [B,4] float32
    float* out = (float*)d_out;               // [B,2] float32

    const int B = in_sizes[0] / 2;
    const int nPairs = B / 2;

    if (nPairs > 0) {
        const int threads = 256;
        const int blocks = (nPairs + threads - 1) / threads;
        quantum_layer_kernel<<<blocks, threads, 0, stream>>>(
            (const v4f*)x, (const v4f*)p, (v4f*)out, nPairs);
    }
    const int done = nPairs * 2;
    if (done < B) {
        const int rem = B - done;
        quantum_layer_tail_kernel<<<(rem + 63) / 64, 64, 0, stream>>>(
            x, p, out, done, B);
    }
}